// BaseCNet_20263655702653
// MI455X (gfx1250) — compile-verified
//
#include <hip/hip_runtime.h>
#include <hip/hip_bf16.h>

typedef float v2f __attribute__((ext_vector_type(2)));
typedef float v8f __attribute__((ext_vector_type(8)));

#define EPS 1e-5f
#define TN 4            // column tiles (of 16) per wave -> 64 cols
#define KCHUNK 64       // K chunk staged in LDS
#define LDS_STRIDE 65   // pad to avoid 64-bank conflicts between K rows

// ---------------------------------------------------------------------------
// WMMA f32 GEMM: Z[nrows x Cout] = X[nrows x K] * W[K x Cout] (+bias) (+accum)
// Wave -> 16x64 output strip (4 WMMA accumulators), W staged through LDS.
// Optional per-column sum/sumsq accumulation for training-mode BatchNorm.
// ---------------------------------------------------------------------------
__global__ void gemm_wmma_f32(const float* __restrict__ X, const float* __restrict__ W,
                              const float* __restrict__ bias, float* __restrict__ Z,
                              int nrows, int K, int Cout, int accum, int stats,
                              float* __restrict__ colsum, float* __restrict__ colsumsq) {
    __shared__ float sw[KCHUNK * LDS_STRIDE];

    const int wave = threadIdx.x >> 5;
    const int lane = threadIdx.x & 31;
    const int nrowTiles = nrows >> 4;               // nrows is a multiple of 16
    int tileM = blockIdx.x * (blockDim.x >> 5) + wave;
    const bool active = tileM < nrowTiles;
    if (!active) tileM = nrowTiles - 1;             // clamp: keep loads in-bounds, skip stores
    const int row0 = tileM * 16;
    const int colBase = blockIdx.y * (TN * 16);

    const int half = lane >> 4;                     // 0: K=0,1 / rows 0-7 ; 1: K=2,3 / rows 8-15
    const int l15  = lane & 15;

    v8f c[TN];
    if (accum) {
#pragma unroll
        for (int t = 0; t < TN; ++t)
#pragma unroll
            for (int j = 0; j < 8; ++j)
                c[t][j] = Z[(long)(row0 + half * 8 + j) * Cout + colBase + t * 16 + l15];
    } else if (bias != nullptr) {
#pragma unroll
        for (int t = 0; t < TN; ++t) {
            const float bv = bias[colBase + t * 16 + l15];
#pragma unroll
            for (int j = 0; j < 8; ++j) c[t][j] = bv;
        }
    } else {
#pragma unroll
        for (int t = 0; t < TN; ++t)
#pragma unroll
            for (int j = 0; j < 8; ++j) c[t][j] = 0.0f;
    }

    const float* xrow = X + (long)(row0 + l15) * K;  // A fragment: lane -> row M=l15

    for (int kb = 0; kb < K; kb += KCHUNK) {
        const int kc = min(KCHUNK, K - kb);
        // cooperative, coalesced stage of W[kb..kb+kc) x 64 cols into LDS
        for (int idx = threadIdx.x; idx < kc * 64; idx += 256) {
            const int r = idx >> 6, cc = idx & 63;
            sw[r * LDS_STRIDE + cc] = W[(long)(kb + r) * Cout + colBase + cc];
        }
        __syncthreads();

        for (int kk = 0; kk < kc; kk += 4) {
            const int k0 = kk + half * 2;
            v2f a;
            a.x = xrow[kb + k0];
            a.y = xrow[kb + k0 + 1];
#pragma unroll
            for (int t = 0; t < TN; ++t) {
                v2f b;
                b.x = sw[k0 * LDS_STRIDE + t * 16 + l15];
                b.y = sw[(k0 + 1) * LDS_STRIDE + t * 16 + l15];
                c[t] = __builtin_amdgcn_wmma_f32_16x16x4_f32(false, a, false, b,
                                                             (short)0, c[t], false, false);
            }
        }
        __syncthreads();
    }

    if (active) {
#pragma unroll
        for (int t = 0; t < TN; ++t) {
            const int col = colBase + t * 16 + l15;
            float s = 0.f, q = 0.f;
#pragma unroll
            for (int j = 0; j < 8; ++j) {
                const float v = c[t][j];
                Z[(long)(row0 + half * 8 + j) * Cout + col] = v;
                s += v; q += v * v;
            }
            if (stats) {
                s += __shfl_down(s, 16, 32);   // merge the two half-wave partials per column
                q += __shfl_down(q, 16, 32);
                if (half == 0) {
                    atomicAdd(&colsum[col], s);
                    atomicAdd(&colsumsq[col], q);
                }
            }
        }
    }
}

// ---------------------------------------------------------------------------
// Graph helpers
// ---------------------------------------------------------------------------
__global__ void fill_kernel(float* __restrict__ p, long n, float v) {
    long i = blockIdx.x * (long)blockDim.x + threadIdx.x;
    if (i < n) p[i] = v;
}

__global__ void deg_kernel(const int* __restrict__ dst, float* __restrict__ deg, int E) {
    int i = blockIdx.x * blockDim.x + threadIdx.x;
    if (i < E) atomicAdd(&deg[dst[i]], 1.0f);
}

__global__ void rsqrt_inplace(float* __restrict__ p, int n) {
    int i = blockIdx.x * blockDim.x + threadIdx.x;
    if (i < n) p[i] = rsqrtf(p[i]);
}

// agg[dst] += h[src] * dinv[src]*dinv[dst]; edges e<E from lists, e>=E are self-loops.
__global__ void scatter_kernel(const float* __restrict__ H, const float* __restrict__ dinv,
                               const int* __restrict__ src, const int* __restrict__ dst,
                               float* __restrict__ agg, int E, int n, int C) {
    const int cq = C >> 2;
    long gid = blockIdx.x * (long)blockDim.x + threadIdx.x;
    long total = (long)(E + n) * cq;
    if (gid >= total) return;
    const int e = (int)(gid / cq);
    const int c = (int)(gid % cq) << 2;
    int s, d;
    if (e < E) { s = src[e]; d = dst[e]; } else { s = d = e - E; }
    const float nrm = dinv[s] * dinv[d];
    const float* hp = H + (long)s * C + c;
    float* ap = agg + (long)d * C + c;
    atomicAdd(ap + 0, hp[0] * nrm);
    atomicAdd(ap + 1, hp[1] * nrm);
    atomicAdd(ap + 2, hp[2] * nrm);
    atomicAdd(ap + 3, hp[3] * nrm);
}

// In-place bias add + per-column sum/sumsq (block: 64 cols x 4 row-threads, 1024-row chunks)
__global__ void col_bias_stats(float* __restrict__ Z, const float* __restrict__ bias,
                               float* __restrict__ colsum, float* __restrict__ colsumsq,
                               int nrows, int C) {
    __shared__ float ssum[256], ssq[256];
    const int cx = threadIdx.x & 63;
    const int ry = threadIdx.x >> 6;
    const int col = blockIdx.x * 64 + cx;
    const float bv = bias[col];
    const int r0 = blockIdx.y * 1024;
    const int r1 = min(r0 + 1024, nrows);
    float s = 0.f, q = 0.f;
    for (int r = r0 + ry; r < r1; r += 4) {
        const long idx = (long)r * C + col;
        const float v = Z[idx] + bv;
        Z[idx] = v;
        s += v; q += v * v;
    }
    ssum[threadIdx.x] = s; ssq[threadIdx.x] = q;
    __syncthreads();
    if (ry == 0) {
        s = ssum[cx] + ssum[cx + 64] + ssum[cx + 128] + ssum[cx + 192];
        q = ssq[cx]  + ssq[cx + 64]  + ssq[cx + 128]  + ssq[cx + 192];
        atomicAdd(&colsum[col], s);
        atomicAdd(&colsumsq[col], q);
    }
}

__global__ void finalize_stats(const float* __restrict__ colsum, const float* __restrict__ colsumsq,
                               const float* __restrict__ g, const float* __restrict__ be,
                               float* __restrict__ scale, float* __restrict__ shift,
                               int C, float invN) {
    int c = blockIdx.x * blockDim.x + threadIdx.x;
    if (c >= C) return;
    const float m  = colsum[c] * invN;
    const float vv = colsumsq[c] * invN - m * m;
    const float sc = g[c] * rsqrtf(vv + EPS);
    scale[c] = sc;
    shift[c] = be[c] - m * sc;
}

// out = relu(z*scale + shift); optional fused mean-pool accumulation by graph id
__global__ void apply_bn_relu(const float* __restrict__ Z, const float* __restrict__ scale,
                              const float* __restrict__ shift, float* __restrict__ out,
                              int nrows, int C, const int* __restrict__ batch,
                              float* __restrict__ pool, int doPool) {
    long i = blockIdx.x * (long)blockDim.x + threadIdx.x;
    long total = (long)nrows * C;
    if (i >= total) return;
    const int c = (int)(i % C);
    float v = fmaf(Z[i], scale[c], shift[c]);
    v = v > 0.f ? v : 0.f;
    out[i] = v;
    if (doPool) {
        const int r = (int)(i / C);
        atomicAdd(&pool[batch[r] * C + c], v);
    }
}

__global__ void count_kernel(const int* __restrict__ batch, float* __restrict__ cnt, int n) {
    __shared__ float lc[16];
    if (threadIdx.x < 16) lc[threadIdx.x] = 0.f;
    __syncthreads();
    int i = blockIdx.x * blockDim.x + threadIdx.x;
    if (i < n) atomicAdd(&lc[batch[i]], 1.0f);
    __syncthreads();
    if (threadIdx.x < 16) atomicAdd(&cnt[threadIdx.x], lc[threadIdx.x]);
}

__global__ void pool_mean_kernel(float* __restrict__ pool, const float* __restrict__ cnt) {
    int i = blockIdx.x * blockDim.x + threadIdx.x;
    if (i >= 16 * 256) return;
    pool[i] /= fmaxf(cnt[i / 256], 1.0f);
}

// dec1: z = xg(16x256) @ W(256x512) + b ; BN over 16 rows ; relu   (one thread per column)
__global__ void dec1_kernel(const float* __restrict__ xg, const float* __restrict__ W,
                            const float* __restrict__ b, const float* __restrict__ g,
                            const float* __restrict__ be, float* __restrict__ out) {
    int c = blockIdx.x * blockDim.x + threadIdx.x;
    if (c >= 512) return;
    float z[16];
    for (int gg = 0; gg < 16; ++gg) {
        float acc = b[c];
        for (int k = 0; k < 256; ++k)
            acc = fmaf(xg[gg * 256 + k], W[(long)k * 512 + c], acc);
        z[gg] = acc;
    }
    float m = 0.f;
#pragma unroll
    for (int gg = 0; gg < 16; ++gg) m += z[gg];
    m *= (1.0f / 16.0f);
    float var = 0.f;
#pragma unroll
    for (int gg = 0; gg < 16; ++gg) { const float d = z[gg] - m; var += d * d; }
    var *= (1.0f / 16.0f);
    const float sc = g[c] * rsqrtf(var + EPS);
    const float sh = be[c] - m * sc;
#pragma unroll
    for (int gg = 0; gg < 16; ++gg)
        out[gg * 512 + c] = fmaxf(fmaf(z[gg], sc, sh), 0.f);
}

// dec0: out(16x6000) = g1(16x512) @ W(512x6000) + b
__global__ void dec0_kernel(const float* __restrict__ g1, const float* __restrict__ W,
                            const float* __restrict__ b, float* __restrict__ out) {
    int i = blockIdx.x * blockDim.x + threadIdx.x;
    if (i >= 16 * 6000) return;
    const int g = i / 6000, o = i % 6000;
    float acc = b[o];
    for (int k = 0; k < 512; ++k)
        acc = fmaf(g1[g * 512 + k], W[(long)k * 6000 + o], acc);
    out[i] = acc;
}

// ---------------------------------------------------------------------------
extern "C" void kernel_launch(void* const* d_in, const int* in_sizes, int n_in,
                              void* d_out, int out_size, void* d_ws, size_t ws_size,
                              hipStream_t stream) {
    const int C_IN = 8;
    const int n = in_sizes[0] / C_IN;          // 100000
    const int E = in_sizes[1] / 2;             // 800000

    const float* x   = (const float*)d_in[0];
    const int* eidx  = (const int*)d_in[1];
    const int* src   = eidx;
    const int* dst   = eidx + E;
    const int* batch = (const int*)d_in[2];

    // params in setup_inputs() insertion order
    const float* P[40];
    for (int i = 3; i < n_in && i < 40; ++i) P[i] = (const float*)d_in[i];
    // enc0: 3..10  enc1: 11..18  enc2: 19..26  lin: 27..30  dec1: 31..34  dec0: 35..36

    float* ws = (float*)d_ws;
    size_t o = 0;
    float* dinv = ws + o;  o += (size_t)n;
    float* A    = ws + o;  o += (size_t)n * 256;   // GEMM output / h scratch
    float* B    = ws + o;  o += (size_t)n * 128;   // bn-applied tmp / aggregation scratch
    float* x0   = ws + o;  o += (size_t)n * 64;
    float* x1   = ws + o;  o += (size_t)n * 128;
    float* x2   = ws + o;  o += (size_t)n * 128;
    float* colsum   = ws + o;  o += 512;
    float* colsumsq = ws + o;  o += 512;
    float* scale    = ws + o;  o += 512;
    float* shift    = ws + o;  o += 512;
    float* pool = ws + o;  o += 16 * 256;
    float* cnt  = ws + o;  o += 16;
    float* g1b  = ws + o;  o += 16 * 512;
    (void)ws_size;

    const float invN = 1.0f / (float)n;

    auto run_gemm = [&](const float* X, const float* W, const float* bias, float* Z,
                        int K, int Cout, int accum, int stats) {
        dim3 grid((n / 16 + 7) / 8, Cout / 64);
        gemm_wmma_f32<<<grid, 256, 0, stream>>>(X, W, bias, Z, n, K, Cout,
                                                accum, stats, colsum, colsumsq);
    };

    // symmetric-norm degrees: deg = 1 (self loop) + in-edges ; dinv = rsqrt(deg)
    fill_kernel<<<(n + 255) / 256, 256, 0, stream>>>(dinv, n, 1.0f);
    deg_kernel<<<(E + 255) / 256, 256, 0, stream>>>(dst, dinv, E);
    rsqrt_inplace<<<(n + 255) / 256, 256, 0, stream>>>(dinv, n);

    auto encoder = [&](const float* xin, int Cin, int Cout, const float* const* pp, float* xout) {
        const float *w1 = pp[0], *b1 = pp[1], *gm = pp[2], *bm = pp[3];
        const float *wc = pp[4], *bc = pp[5], *gc = pp[6], *bce = pp[7];
        const long tot = (long)n * Cout;
        // MLP: z = x@w1+b1 with fused column stats
        hipMemsetAsync(colsum, 0, 1024 * sizeof(float), stream);
        run_gemm(xin, w1, b1, A, Cin, Cout, 0, 1);
        finalize_stats<<<(Cout + 255) / 256, 256, 0, stream>>>(colsum, colsumsq, gm, bm,
                                                               scale, shift, Cout, invN);
        apply_bn_relu<<<(tot + 255) / 256, 256, 0, stream>>>(A, scale, shift, B, n, Cout,
                                                             nullptr, nullptr, 0);
        // GCN: h = t@wc ; agg[dst] += h[src]*norm ; +bc ; BN ; relu
        run_gemm(B, wc, nullptr, A, Cout, Cout, 0, 0);
        hipMemsetAsync(B, 0, (size_t)n * Cout * sizeof(float), stream);
        hipMemsetAsync(colsum, 0, 1024 * sizeof(float), stream);
        const long sct = (long)(E + n) * (Cout / 4);
        scatter_kernel<<<(sct + 255) / 256, 256, 0, stream>>>(A, dinv, src, dst, B, E, n, Cout);
        col_bias_stats<<<dim3(Cout / 64, (n + 1023) / 1024), 256, 0, stream>>>(B, bc, colsum,
                                                                               colsumsq, n, Cout);
        finalize_stats<<<(Cout + 255) / 256, 256, 0, stream>>>(colsum, colsumsq, gc, bce,
                                                               scale, shift, Cout, invN);
        apply_bn_relu<<<(tot + 255) / 256, 256, 0, stream>>>(B, scale, shift, xout, n, Cout,
                                                             nullptr, nullptr, 0);
    };

    encoder(x,  C_IN, 64,  &P[3],  x0);
    encoder(x0, 64,   128, &P[11], x1);
    encoder(x1, 128,  128, &P[19], x2);

    // lin: relu(bn([x0|x1|x2] @ W(320x256) + b)) with fused mean-pool
    hipMemsetAsync(colsum, 0, 1024 * sizeof(float), stream);
    run_gemm(x0, P[27],             P[28], A, 64,  256, 0, 0);
    run_gemm(x1, P[27] + 64 * 256,  nullptr, A, 128, 256, 1, 0);
    run_gemm(x2, P[27] + 192 * 256, nullptr, A, 128, 256, 1, 1);
    finalize_stats<<<1, 256, 0, stream>>>(colsum, colsumsq, P[29], P[30], scale, shift, 256, invN);
    hipMemsetAsync(pool, 0, (16 * 256 + 16) * sizeof(float), stream);   // pool + cnt contiguous
    count_kernel<<<(n + 255) / 256, 256, 0, stream>>>(batch, cnt, n);
    {
        const long tot = (long)n * 256;
        apply_bn_relu<<<(tot + 255) / 256, 256, 0, stream>>>(A, scale, shift, A, n, 256,
                                                             batch, pool, 1);
    }
    pool_mean_kernel<<<16, 256, 0, stream>>>(pool, cnt);

    // decoder head
    dec1_kernel<<<2, 256, 0, stream>>>(pool, P[31], P[32], P[33], P[34], g1b);
    dec0_kernel<<<(16 * 6000 + 255) / 256, 256, 0, stream>>>(g1b, P[35], P[36], (float*)d_out);
}